// BlockBlastValueNet1PmultikernelFlattenned_32890859552801
// MI455X (gfx1250) — compile-verified
//
#include <hip/hip_runtime.h>

typedef __attribute__((ext_vector_type(2))) float v2f;
typedef __attribute__((ext_vector_type(8))) float v8f;

// ---------------------------------------------------------------------------
// Workspace layout (floats):
//   G  : [128 x 64]  folded (conv+W1) per-branch matrices, row-major   @ 0
//   g  : [128]       folded biases                                     @ 8192
//   M  : [64 x 128]  folded (W2 -> fc_w1) matrix, row-major            @ 8320
//   m  : [64]        folded fc1 bias                                   @ 16512
// total 16576 floats = 66304 bytes
// ---------------------------------------------------------------------------
#define WS_G 0
#define WS_g 8192
#define WS_M 8320
#define WS_m 16512

struct Params {
  const float* cw[8];
  const float* cb[8];
  const float* w1[8];
  const float* b1[8];
  const float* w2[8];
  const float* b2[8];
  const float* fc_w1;
  const float* fc_b1;
};

// One block per branch (0..7) folds conv+dense1 -> G (16x64) + g (16).
// Block 8 folds dense2+fc1 -> M (64x128) + m (64).
__global__ void precompute_fold(Params P, float* __restrict__ ws) {
  const int KH[8] = {1, 2, 3, 4, 5, 8, 1, 8};
  const int KW[8] = {1, 2, 3, 4, 5, 8, 8, 1};
  const int FS[8] = {1, 6, 8, 8, 16, 32, 4, 4};
  const int PH[8] = {0, 1, 1, 2, 2, 0, 0, 0};
  const int PW[8] = {0, 1, 1, 2, 2, 0, 0, 0};

  float* G  = ws + WS_G;
  float* g  = ws + WS_g;
  float* Mw = ws + WS_M;
  float* mv = ws + WS_m;

  const int b = blockIdx.x;
  if (b < 8) {
    __shared__ __align__(16) float Gs[16 * 64];
    const int o = threadIdx.x;
    if (o < 16) {
      for (int q = 0; q < 64; ++q) Gs[o * 64 + q] = 0.0f;
      const int kh = KH[b], kw = KW[b], fs = FS[b], ph = PH[b], pw = PW[b];
      const int OH = 8 + 2 * ph - kh + 1;
      const int OW = 8 + 2 * pw - kw + 1;
      const int n  = fs * OH * OW;
      const float* cw = P.cw[b];
      const float* cb = P.cb[b];
      const float* w1 = P.w1[b];
      float gacc = P.b1[b][o];
      for (int c = 0; c < fs; ++c) {
        for (int oy = 0; oy < OH; ++oy) {
          for (int ox = 0; ox < OW; ++ox) {
            const float w1v = w1[o * n + (c * OH + oy) * OW + ox];
            gacc += cb[c] * w1v;
            for (int ky = 0; ky < kh; ++ky) {
              for (int kx = 0; kx < kw; ++kx) {
                const float wgt = w1v * cw[(c * kh + ky) * kw + kx];
                const int by = oy + ky - ph;
                const int bx = ox + kx - pw;
                if (by >= 0 && by < 8 && bx >= 0 && bx < 8)
                  Gs[o * 64 + by * 8 + bx] += wgt;
                else
                  gacc += wgt;   // pad value is 1.0 -> constant contribution
              }
            }
          }
        }
      }
      for (int q = 0; q < 64; ++q) G[(b * 16 + o) * 64 + q] = Gs[o * 64 + q];
      g[b * 16 + o] = gacc;
    }
  } else {
    const int tid = threadIdx.x;
    // M[r][i*16+j] = sum_q fc_w1[r, 8i+q] * w2_i[q, j]
    for (int e = tid; e < 64 * 128; e += blockDim.x) {
      const int r = e >> 7;
      const int col = e & 127;
      const int i = col >> 4;
      const int j = col & 15;
      float s = 0.0f;
      for (int q = 0; q < 8; ++q)
        s += P.fc_w1[r * 64 + i * 8 + q] * P.w2[i][q * 16 + j];
      Mw[e] = s;
    }
    if (tid < 64) {
      float s = P.fc_b1[tid];
      for (int i = 0; i < 8; ++i)
        for (int q = 0; q < 8; ++q)
          s += P.fc_w1[tid * 64 + i * 8 + q] * P.b2[i][q];
      mv[tid] = s;
    }
  }
}

__device__ __forceinline__ float leaky(float v) {
  return v > 0.0f ? v : 0.01f * v;
}

__device__ __forceinline__ v8f wmma_f32x4(v2f a, v2f b, v8f c) {
  return __builtin_amdgcn_wmma_f32_16x16x4_f32(false, a, false, b, (short)0, c,
                                               false, false);
}

// 4 waves / block, 16 samples per wave -> 64 samples / block.
// Fragment layouts (ISA 7.12.2, wave32):
//   A 16x4 f32 : lane holds (m = lane&15, k = 2*(lane>>4) + {0,1}) -> v2f
//   B 4x16 f32 : lane holds (n = lane&15, k = 2*(lane>>4) + {0,1}) -> v2f
//   C/D 16x16  : lane holds (n = lane&15, m = 8*(lane>>4) + j), j = vgpr 0..7
#define FWD_WAVES 4
__global__ __launch_bounds__(FWD_WAVES * 32) void forward_fused(
    const float* __restrict__ board, const float* __restrict__ ws,
    const float* __restrict__ fcw2, const float* __restrict__ fcb2,
    const float* __restrict__ fcw3, const float* __restrict__ fcb3,
    float* __restrict__ out) {
  const float* G  = ws + WS_G;
  const float* g  = ws + WS_g;
  const float* Mw = ws + WS_M;
  const float* mv = ws + WS_m;

  __shared__ __align__(16) float lds_h[FWD_WAVES * 16 * 128];
  __shared__ __align__(16) float lds_z[FWD_WAVES * 16 * 64];

  const int wave = threadIdx.x >> 5;
  const int lane = threadIdx.x & 31;
  const int half = lane >> 4;   // K-half / M-half selector
  const int lr   = lane & 15;   // A row / B column (= sample slot)
  const int s0   = blockIdx.x * (FWD_WAVES * 16) + wave * 16;
  const int smp  = s0 + lr;     // this lane's sample for B/C fragments

  float* hW = lds_h + wave * 16 * 128;  // [sample][128] row-major
  float* zW = lds_z + wave * 16 * 64;   // [sample][64]  row-major

  // ---- stage 1: u = leaky(G x + g), 128x64 fp32 GEMM via WMMA ----
  const float* Bx = board + (size_t)smp * 64 + half * 2;
  for (int mt = 0; mt < 8; ++mt) {
    v8f acc = {0.f, 0.f, 0.f, 0.f, 0.f, 0.f, 0.f, 0.f};
    const float* Ar = G + (mt * 16 + lr) * 64 + half * 2;
#pragma unroll
    for (int kt = 0; kt < 16; ++kt) {
      v2f a = *(const v2f*)(Ar + kt * 4);
      v2f b = *(const v2f*)(Bx + kt * 4);
      acc = wmma_f32x4(a, b, acc);
    }
    const int mbase = mt * 16 + half * 8;
#pragma unroll
    for (int j = 0; j < 8; ++j) {
      hW[lr * 128 + mbase + j] = leaky(acc[j] + g[mbase + j]);
    }
  }
  asm volatile("s_wait_dscnt 0" ::: "memory");

  // ---- stage 2: z = leaky(M u + m), 64x128 fp32 GEMM via WMMA ----
  const float* Bh = hW + lr * 128 + half * 2;
  for (int mt = 0; mt < 4; ++mt) {
    v8f acc = {0.f, 0.f, 0.f, 0.f, 0.f, 0.f, 0.f, 0.f};
    const float* Ar = Mw + (mt * 16 + lr) * 128 + half * 2;
#pragma unroll
    for (int kt = 0; kt < 32; ++kt) {
      v2f a = *(const v2f*)(Ar + kt * 4);
      v2f b = *(const v2f*)(Bh + kt * 4);
      acc = wmma_f32x4(a, b, acc);
    }
    const int mbase = mt * 16 + half * 8;
#pragma unroll
    for (int j = 0; j < 8; ++j) {
      zW[lr * 64 + mbase + j] = leaky(acc[j] + mv[mbase + j]);
    }
  }
  asm volatile("s_wait_dscnt 0" ::: "memory");

  // ---- stage 3: t = leaky(fc_w2 z + fc_b2), 16x64 fp32 GEMM via WMMA ----
  {
    v8f acc = {0.f, 0.f, 0.f, 0.f, 0.f, 0.f, 0.f, 0.f};
    const float* Ar = fcw2 + lr * 64 + half * 2;
    const float* Bz = zW + lr * 64 + half * 2;
#pragma unroll
    for (int kt = 0; kt < 16; ++kt) {
      v2f a = *(const v2f*)(Ar + kt * 4);
      v2f b = *(const v2f*)(Bz + kt * 4);
      acc = wmma_f32x4(a, b, acc);
    }
    // ---- stage 4: out = fc_w3 . t + fc_b3 (per-lane dot + half combine) ----
    const int mbase = half * 8;
    float p = 0.0f;
#pragma unroll
    for (int j = 0; j < 8; ++j) {
      p += fcw3[mbase + j] * leaky(acc[j] + fcb2[mbase + j]);
    }
    const float tot = p + __shfl_xor(p, 16, 32);
    if (half == 0) {
      out[smp] = tot + fcb3[0];
    }
  }
}

extern "C" void kernel_launch(void* const* d_in, const int* in_sizes, int n_in,
                              void* d_out, int out_size, void* d_ws,
                              size_t ws_size, hipStream_t stream) {
  (void)in_sizes; (void)n_in; (void)out_size; (void)ws_size;

  const float* board = (const float*)d_in[0];
  Params P;
  for (int i = 0; i < 8; ++i) {
    P.cw[i] = (const float*)d_in[1 + 6 * i];
    P.cb[i] = (const float*)d_in[2 + 6 * i];
    P.w1[i] = (const float*)d_in[3 + 6 * i];
    P.b1[i] = (const float*)d_in[4 + 6 * i];
    P.w2[i] = (const float*)d_in[5 + 6 * i];
    P.b2[i] = (const float*)d_in[6 + 6 * i];
  }
  P.fc_w1 = (const float*)d_in[49];
  P.fc_b1 = (const float*)d_in[50];
  const float* fcw2 = (const float*)d_in[51];
  const float* fcb2 = (const float*)d_in[52];
  const float* fcw3 = (const float*)d_in[53];
  const float* fcb3 = (const float*)d_in[54];

  float* ws = (float*)d_ws;
  float* out = (float*)d_out;

  // Fold conv+dense layers into G/g (128x64) and M/m (64x128) once per launch.
  precompute_fold<<<9, 256, 0, stream>>>(P, ws);

  // 65536 samples / (4 waves * 16 samples) = 1024 blocks.
  const int BATCH = 65536;
  const int per_block = FWD_WAVES * 16;
  forward_fused<<<BATCH / per_block, FWD_WAVES * 32, 0, stream>>>(
      board, ws, fcw2, fcb2, fcw3, fcb3, out);
}